// RBS_Gate_state_vector_31963146617520
// MI455X (gfx1250) — compile-verified
//
#include <hip/hip_runtime.h>
#include <math.h>

typedef __attribute__((ext_vector_type(2))) float v2f;
typedef __attribute__((ext_vector_type(4))) float v4f;
typedef __attribute__((ext_vector_type(8))) float v8f;

#define NQ 8192      // state dimension (K and M)
#define BQ 256       // batch (output columns)
#define MB 32        // M rows per workgroup (2 WMMA M-tiles -> 2x B reuse)
#define KC 64        // K-chunk staged per LDS buffer
#define NCHUNK (NQ / KC)
#define FRAG_FLOATS (KC / 4 * 2 * 64)   // 16 ks * 2 mtiles * 64 floats = 2048

// One workgroup (256 threads = 8 wave32) computes a 32-row x 256-col output
// slab. U = cos*U0 + sin*U1 + U2 is fused at stage time (never materialized):
// each 32xKC tile of U0/U1/U2 is read from HBM exactly once, combined, and
// written to LDS pre-packed in V_WMMA_F32_16X16X4_F32 A-fragment order
// ([ks][mtile][lane][2]) so every wave gets A via one conflict-free
// ds_load_b64. Each wave owns 32 columns: 4 accumulators (2 M x 2 N tiles),
// so every B fragment (from L2-resident x) feeds two WMMAs.
__global__ __launch_bounds__(256)
void rbs_gemm_kernel(const float* __restrict__ angle,
                     const float* __restrict__ U0,
                     const float* __restrict__ U1,
                     const float* __restrict__ U2,
                     const float* __restrict__ x,
                     float*       __restrict__ out)
{
    __shared__ float lds[2 * FRAG_FLOATS];   // 2 x 8KB double buffer

    const int tid   = threadIdx.x;
    const int lane  = tid & 31;
    const int wave  = tid >> 5;
    const int mbase = blockIdx.x * MB;

    const float ang = angle[0];
    const float cv  = cosf(ang);
    const float sv  = sinf(ang);

    // ---- staging role: thread (srow, scg) loads two float4 of each U tile ----
    const int srow = tid & 31;            // row within 32-row slab
    const int scg  = (tid >> 5) << 3;     // k offset within chunk: 0,8,...,56
    const int arow = (mbase + srow) * NQ; // < 2^27, fits int
    const int smt  = srow >> 4;           // which M-tile this row feeds
    const int sl2  = (srow & 15) * 2;     // lane-pair slot inside fragment

    // ---- compute role: this wave's 32x32 output slab ----
    const int col0 = wave * 32 + (lane & 15);
    const int krow = (lane >> 4) << 1;    // B half-wave K split: +0 / +2

    v8f c00 = {}, c01 = {};               // M-tile 0, N-tiles 0/1
    v8f c10 = {}, c11 = {};               // M-tile 1, N-tiles 0/1

    // fuse + scatter one staged chunk (two float4 per U matrix) into a buffer
    auto stage_store = [&](int buf, v4f a0, v4f a1, v4f b0, v4f b1,
                           v4f e0, v4f e1) {
        v4f u0 = a0 * cv + b0 * sv + e0;  // k = scg .. scg+3
        v4f u1 = a1 * cv + b1 * sv + e1;  // k = scg+4 .. scg+7
        const int f0 = buf * FRAG_FLOATS + (((scg >> 2) * 2 + smt) * 64) + sl2;
        const int f1 = f0 + 2 * 64;       // next ks fragment
        *(v2f*)(lds + f0)      = (v2f){u0.x, u0.y};  // lane = l      (K0,K1)
        *(v2f*)(lds + f0 + 32) = (v2f){u0.z, u0.w};  // lane = l + 16 (K2,K3)
        *(v2f*)(lds + f1)      = (v2f){u1.x, u1.y};
        *(v2f*)(lds + f1 + 32) = (v2f){u1.z, u1.w};
    };

    // one chunk of WMMAs from buffer `buf`, global K base `kcg`
    auto compute = [&](int buf, int kcg) {
        const float* lbuf = lds + buf * FRAG_FLOATS;
#pragma unroll
        for (int ks = 0; ks < KC / 4; ++ks) {
            v2f a0 = *(const v2f*)(lbuf + (ks * 2 + 0) * 64 + lane * 2);
            v2f a1 = *(const v2f*)(lbuf + (ks * 2 + 1) * 64 + lane * 2);

            const float* xb = x + (kcg + ks * 4 + krow) * BQ + col0;
            v2f b0 = (v2f){xb[0],  xb[BQ]};
            v2f b1 = (v2f){xb[16], xb[16 + BQ]};

            c00 = __builtin_amdgcn_wmma_f32_16x16x4_f32(
                      false, a0, false, b0, (short)0, c00, false, false);
            c01 = __builtin_amdgcn_wmma_f32_16x16x4_f32(
                      false, a0, false, b1, (short)0, c01, false, false);
            c10 = __builtin_amdgcn_wmma_f32_16x16x4_f32(
                      false, a1, false, b0, (short)0, c10, false, false);
            c11 = __builtin_amdgcn_wmma_f32_16x16x4_f32(
                      false, a1, false, b1, (short)0, c11, false, false);
        }
    };

    // ---------- prologue: stage chunk 0 into buffer 0 ----------
    {
        const float* g0 = U0 + arow + scg;
        const float* g1 = U1 + arow + scg;
        const float* g2 = U2 + arow + scg;
        stage_store(0, *(const v4f*)g0, *(const v4f*)(g0 + 4),
                       *(const v4f*)g1, *(const v4f*)(g1 + 4),
                       *(const v4f*)g2, *(const v4f*)(g2 + 4));
    }
    __syncthreads();

    // ---------- steady state: prefetch i+1, compute i, stage i+1 ----------
    int cur = 0;
    for (int i = 0; i < NCHUNK - 1; ++i) {
        // L2 prefetch two chunks ahead (global_prefetch_b8): gives HBM a full
        // extra chunk of lead time beyond the register-level prefetch below.
        {
            int ip = i + 2;
            if (ip > NCHUNK - 1) ip = NCHUNK - 1;   // stay inside allocation
            const int kp = ip * KC + scg;
            __builtin_prefetch(U0 + arow + kp, 0, 3);
            __builtin_prefetch(U1 + arow + kp, 0, 3);
            __builtin_prefetch(U2 + arow + kp, 0, 3);
        }

        const int kn = (i + 1) * KC + scg;
        const float* g0 = U0 + arow + kn;
        const float* g1 = U1 + arow + kn;
        const float* g2 = U2 + arow + kn;
        v4f q00 = *(const v4f*)g0, q01 = *(const v4f*)(g0 + 4);
        v4f q10 = *(const v4f*)g1, q11 = *(const v4f*)(g1 + 4);
        v4f q20 = *(const v4f*)g2, q21 = *(const v4f*)(g2 + 4);

        compute(cur, i * KC);

        stage_store(cur ^ 1, q00, q01, q10, q11, q20, q21);
        __syncthreads();
        cur ^= 1;
    }

    // ---------- final chunk ----------
    compute(cur, (NCHUNK - 1) * KC);

    // ---------- epilogue: D layout VGPR r -> M = r (+8 for lanes 16-31) ----------
    const int mrow = mbase + ((lane >> 4) << 3);
#pragma unroll
    for (int r = 0; r < 8; ++r) {
        out[(mrow + r) * BQ + col0]           = c00[r];
        out[(mrow + r) * BQ + col0 + 16]      = c01[r];
        out[(mrow + 16 + r) * BQ + col0]      = c10[r];
        out[(mrow + 16 + r) * BQ + col0 + 16] = c11[r];
    }
}

extern "C" void kernel_launch(void* const* d_in, const int* in_sizes, int n_in,
                              void* d_out, int out_size, void* d_ws, size_t ws_size,
                              hipStream_t stream)
{
    const float* angle = (const float*)d_in[0];
    const float* U0    = (const float*)d_in[1];
    const float* U1    = (const float*)d_in[2];
    const float* U2    = (const float*)d_in[3];
    const float* x     = (const float*)d_in[4];
    float*       out   = (float*)d_out;

    rbs_gemm_kernel<<<dim3(NQ / MB), dim3(256), 0, stream>>>(
        angle, U0, U1, U2, x, out);
}